// RotAttLayer_59322088292475
// MI455X (gfx1250) — compile-verified
//
#include <hip/hip_runtime.h>

#define N_ENT   200000
#define N_REL   500
#define N_TRIP  200000
#define DIM     128
#define K_DIM   384
#define MARGIN_F 6.0f
#define LRELU   0.01f
#define BN_EPS_F 1e-5f
#define REL_RANGE_F (8.0f/128.0f)
#define PI_F 3.14159265358979323846f

typedef __attribute__((ext_vector_type(2))) float v2f;
typedef __attribute__((ext_vector_type(8))) float v8f;

// ---- workspace layout (float offsets) ----
#define WS_ENT_SCALE 0          // 200000
#define WS_REL_SCALE 200000     // 512
#define WS_SA        200512     // 128  sum of e0+e1 per col
#define WS_QA        200640     // 128  sum of e0^2+e1^2 per col
#define WS_QR        200768     // 128  sum of er^2 per col
#define WS_WB        200896     // 384*128 folded weights, WMMA-ready packing
#define WS_BIAS2     250048     // 128
#define WS_SC1       250176     // 128  sum c
#define WS_SC2       250304     // 128  sum c^2
#define WS_A1        250432     // 128
#define WS_B1N       250560     // 128
#define WS_EBS       250688     // 200000
#define WS_CNT       450688     // 512
#define WS_TOTAL     451200

#define OUT_HENT  0
#define OUT_HREL  (N_ENT*DIM)
#define OUT_SCORE (N_ENT*DIM + N_REL*DIM)

__global__ void zero_kernel(float* p, int n) {
  int stride = gridDim.x * blockDim.x;
  for (int i = blockIdx.x * blockDim.x + threadIdx.x; i < n; i += stride) p[i] = 0.0f;
}

// one wave per row: L2-norm clip scale = min(1, 1/max(||x||,1e-12))
__global__ void scale_kernel(const float* __restrict__ ent, const float* __restrict__ rel,
                             float* __restrict__ ent_scale, float* __restrict__ rel_scale) {
  int wave = (blockIdx.x * blockDim.x + threadIdx.x) >> 5;
  int lane = threadIdx.x & 31;
  if (wave >= N_ENT + N_REL) return;
  const float* src = (wave < N_ENT) ? (ent + (size_t)wave * DIM)
                                    : (rel + (size_t)(wave - N_ENT) * DIM);
  float ss = 0.0f;
  for (int k = lane; k < DIM; k += 32) { float v = src[k]; ss += v * v; }
  for (int off = 16; off > 0; off >>= 1) ss += __shfl_xor(ss, off, 32);
  if (lane == 0) {
    float sc = fminf(1.0f, 1.0f / fmaxf(sqrtf(ss), 1e-12f));
    if (wave < N_ENT) ent_scale[wave] = sc; else rel_scale[wave - N_ENT] = sc;
  }
}

// BN0 statistics: per-column sums over gathered, norm-clipped embeddings. Also rel counts.
__global__ void stats0_kernel(const int* __restrict__ trip,
                              const float* __restrict__ ent, const float* __restrict__ rel,
                              const float* __restrict__ ent_scale, const float* __restrict__ rel_scale,
                              float* __restrict__ Sa, float* __restrict__ Qa, float* __restrict__ Qr,
                              float* __restrict__ cnt) {
  __shared__ float sSa[DIM], sQa[DIM], sQr[DIM];
  int j = threadIdx.x & (DIM - 1);
  int sub = threadIdx.x >> 7;   // two 128-thread groups per block
  if (threadIdx.x < DIM) { sSa[j] = 0.0f; sQa[j] = 0.0f; sQr[j] = 0.0f; }
  __syncthreads();
  float sa = 0.0f, qa = 0.0f, qr = 0.0f;
  for (int i = blockIdx.x * 2 + sub; i < N_TRIP; i += gridDim.x * 2) {
    int t0 = trip[3 * i], t1 = trip[3 * i + 1], t2 = trip[3 * i + 2];
    float s0 = ent_scale[t0], s1 = ent_scale[t1], sr = rel_scale[t2];
    float v0 = ent[(size_t)t0 * DIM + j] * s0;
    float v1 = ent[(size_t)t1 * DIM + j] * s1;
    float vr = rel[(size_t)t2 * DIM + j] * sr;
    sa += v0 + v1; qa += v0 * v0 + v1 * v1; qr += vr * vr;
    if (j == 0) atomicAdd(&cnt[t2], 1.0f);
  }
  atomicAdd(&sSa[j], sa); atomicAdd(&sQa[j], qa); atomicAdd(&sQr[j], qr);
  __syncthreads();
  if (threadIdx.x < DIM) {
    atomicAdd(&Sa[j], sSa[j]); atomicAdd(&Qa[j], sQa[j]); atomicAdd(&Qr[j], sQr[j]);
  }
}

// Fold BN0 into weights, packed for WMMA B-operand b64 loads:
//   element (K, col) with K = 4g + 2h + j  (g = K-group, h = lane-half, j = pair idx)
//   Wb[((g*2 + h)*128 + col)*2 + j] = a0[K] * W_a[col*384 + K]
// bias2[col] = b_a[col] + sum_K b0[K]*W_a[col*384 + K]
__global__ void prep_kernel(const float* __restrict__ W_a, const float* __restrict__ b_a,
                            const float* __restrict__ gamma0, const float* __restrict__ beta0,
                            const float* __restrict__ Sa, const float* __restrict__ Qa,
                            const float* __restrict__ Qr,
                            float* __restrict__ Wb, float* __restrict__ bias2) {
  __shared__ float sa0[K_DIM], sb0[K_DIM];
  int tid = threadIdx.x;
  const float inv2n = 1.0f / (2.0f * (float)N_TRIP);
  for (int jj = tid; jj < K_DIM; jj += blockDim.x) {
    float mu, ex2;
    if (jj < 2 * DIM) { int j = jj & (DIM - 1); mu = Sa[j] * inv2n; ex2 = Qa[j] * inv2n; }
    else              { int j = jj - 2 * DIM;   mu = 0.0f;          ex2 = Qr[j] * (1.0f / (float)N_TRIP); }
    float var = ex2 - mu * mu;
    float a = gamma0[jj] * rsqrtf(var + BN_EPS_F);
    sa0[jj] = a; sb0[jj] = beta0[jj] - mu * a;
  }
  __syncthreads();
  for (int idx = tid; idx < K_DIM * DIM; idx += blockDim.x) {
    int j   = idx & 1;
    int col = (idx >> 1) & (DIM - 1);
    int h   = (idx >> 8) & 1;
    int g   = idx >> 9;
    int K   = 4 * g + 2 * h + j;
    Wb[idx] = sa0[K] * W_a[col * K_DIM + K];
  }
  if (tid < DIM) {
    float acc = b_a[tid];
    for (int j = 0; j < K_DIM; ++j) acc += sb0[j] * W_a[tid * K_DIM + j];
    bias2[tid] = acc;
  }
}

#define AT_STRIDE 388   // 388 % 64 == 4 -> conflict-free 16-row column reads
#define M_TILE   32     // two 16-row WMMA subtiles per wave, shared B register

// Shared WMMA GEMM: 400000x384 @ 384x128, block tile = 32 rows x 128 cols.
// PASS2=false: accumulate per-column sum(c), sum(c^2) for BN1.
// PASS2=true : apply BN1, attention score, exp, and atomic scatter into h_ent/h_rel accumulators.
template<bool PASS2>
__global__ void gemm_kernel(const int* __restrict__ trip,
                            const float* __restrict__ ent, const float* __restrict__ rel,
                            const float* __restrict__ ent_scale, const float* __restrict__ rel_scale,
                            const float* __restrict__ Wb, const float* __restrict__ bias2,
                            const float* __restrict__ a1, const float* __restrict__ b1n,
                            const float* __restrict__ W2, const float* __restrict__ b2s,
                            float* __restrict__ sc1, float* __restrict__ sc2,
                            float* __restrict__ hs, float* __restrict__ ebs,
                            float* __restrict__ hrel) {
  __shared__ float Atile[M_TILE * AT_STRIDE];
  __shared__ int   sTgt[M_TILE], sSec[M_TILE], sRel[M_TILE];
  __shared__ float sSign[M_TILE], sDot[M_TILE], sEb[M_TILE];

  int tid = threadIdx.x;
  int lane = tid & 31;
  int wave = tid >> 5;          // 8 waves -> N-tiles of 16 columns
  int n0 = wave << 4;
  int halfsel = lane >> 4;      // ISA A/B layout: lanes 0-15 hold K={0,1}, 16-31 hold K={2,3}
  int ln = lane & 15;
  int col = n0 + ln;

  float s1acc = 0.0f, s2acc = 0.0f;
  const int numTiles = (2 * N_TRIP) / M_TILE;

  for (int tile = blockIdx.x; tile < numTiles; tile += gridDim.x) {
    int rowBase = tile * M_TILE;
    __syncthreads();
    if (tid < M_TILE) {
      int i = rowBase + tid;
      bool flip = (i >= N_TRIP);
      int i0 = flip ? i - N_TRIP : i;
      int t0 = trip[3 * i0], t1 = trip[3 * i0 + 1], t2 = trip[3 * i0 + 2];
      sTgt[tid]  = flip ? t1 : t0;   // cols 0-127 source AND segment-sum target
      sSec[tid]  = flip ? t0 : t1;   // cols 128-255 source
      sRel[tid]  = t2;
      sSign[tid] = flip ? -1.0f : 1.0f;
      if (PASS2) sDot[tid] = 0.0f;
    }
    __syncthreads();
    // gather + clip-scale into LDS (BN0 folded into Wb/bias2)
    for (int idx = tid; idx < M_TILE * K_DIM; idx += 256) {
      int r = idx / K_DIM; int j = idx - r * K_DIM;
      float v;
      if (j < DIM)          { int e = sTgt[r]; v = ent[(size_t)e * DIM + j] * ent_scale[e]; }
      else if (j < 2 * DIM) { int e = sSec[r]; v = ent[(size_t)e * DIM + (j - DIM)] * ent_scale[e]; }
      else                  { int rr = sRel[r]; v = sSign[r] * rel[(size_t)rr * DIM + (j - 2 * DIM)] * rel_scale[rr]; }
      Atile[r * AT_STRIDE + j] = v;
    }
    __syncthreads();

    v8f acc0 = {}, acc1 = {};
    #pragma unroll 4
    for (int k0 = 0; k0 < K_DIM; k0 += 4) {
      int kk = k0 + (halfsel << 1);
      v2f a0, a1v, b;
      a0.x  = Atile[ln * AT_STRIDE + kk];
      a0.y  = Atile[ln * AT_STRIDE + kk + 1];
      a1v.x = Atile[(16 + ln) * AT_STRIDE + kk];
      a1v.y = Atile[(16 + ln) * AT_STRIDE + kk + 1];
      b = *(const v2f*)&Wb[(((k0 >> 2) * 2 + halfsel) * DIM + col) * 2];
      acc0 = __builtin_amdgcn_wmma_f32_16x16x4_f32(false, a0,  false, b, (short)0, acc0, false, false);
      acc1 = __builtin_amdgcn_wmma_f32_16x16x4_f32(false, a1v, false, b, (short)0, acc1, false, false);
    }

    float bb = bias2[col];
    if constexpr (!PASS2) {
      #pragma unroll
      for (int v = 0; v < 8; ++v) {
        float c0 = acc0[v] + bb; s1acc += c0; s2acc += c0 * c0;
        float c1 = acc1[v] + bb; s1acc += c1; s2acc += c1 * c1;
      }
    } else {
      float A1 = a1[col], B1 = b1n[col], w2 = W2[col];
      float cv0[8], cv1[8];
      #pragma unroll
      for (int v = 0; v < 8; ++v) {
        float c0 = (acc0[v] + bb) * A1 + B1;
        float c1 = (acc1[v] + bb) * A1 + B1;
        cv0[v] = c0; cv1[v] = c1;
        atomicAdd(&sDot[v + 8 * halfsel],      c0 * w2);   // per-row attention logit partials
        atomicAdd(&sDot[16 + v + 8 * halfsel], c1 * w2);
      }
      __syncthreads();
      if (tid < M_TILE) {
        float s = sDot[tid] + b2s[0];
        float bneg = -((s >= 0.0f) ? s : LRELU * s);
        float eb = expf(bneg);
        sEb[tid] = eb;
        atomicAdd(&ebs[sTgt[tid]], eb);
      }
      __syncthreads();
      #pragma unroll
      for (int v = 0; v < 8; ++v) {
        int r0 = v + 8 * halfsel;
        int r1 = 16 + r0;
        float t0 = sEb[r0] * cv0[v];
        float t1 = sEb[r1] * cv1[v];
        atomicAdd(&hs[(size_t)sTgt[r0] * DIM + col], t0);
        atomicAdd(&hrel[(size_t)sRel[r0] * DIM + col], sSign[r0] * t0);
        atomicAdd(&hs[(size_t)sTgt[r1] * DIM + col], t1);
        atomicAdd(&hrel[(size_t)sRel[r1] * DIM + col], sSign[r1] * t1);
      }
    }
  }
  if constexpr (!PASS2) {
    atomicAdd(&sc1[col], s1acc);
    atomicAdd(&sc2[col], s2acc);
  }
}

__global__ void bn1fin_kernel(const float* __restrict__ sc1, const float* __restrict__ sc2,
                              const float* __restrict__ gamma1, const float* __restrict__ beta1,
                              float* __restrict__ a1, float* __restrict__ b1n) {
  int k = threadIdx.x;
  if (k < DIM) {
    const float inv2n = 1.0f / (2.0f * (float)N_TRIP);
    float mu = sc1[k] * inv2n;
    float var = sc2[k] * inv2n - mu * mu;
    float a = gamma1[k] * rsqrtf(var + BN_EPS_F);
    a1[k] = a; b1n[k] = beta1[k] - mu * a;
  }
}

__global__ void finalize_kernel(float* __restrict__ h_ent, const float* __restrict__ ebs,
                                float* __restrict__ h_rel, const float* __restrict__ cnt) {
  int stride = gridDim.x * blockDim.x;
  int g = blockIdx.x * blockDim.x + threadIdx.x;
  for (int idx = g; idx < N_ENT * DIM; idx += stride) {
    float d = ebs[idx >> 7];
    d = (d == 0.0f) ? 1e-12f : d;
    h_ent[idx] = h_ent[idx] / d;
  }
  for (int idx = g; idx < N_REL * DIM; idx += stride) {
    float c = fmaxf(cnt[idx >> 7], 1.0f);
    h_rel[idx] = h_rel[idx] / c;
  }
}

// RotatE score, one wave per triplet (uses RAW embeddings per reference)
__global__ void score_kernel(const int* __restrict__ trip,
                             const float* __restrict__ ent, const float* __restrict__ rel,
                             float* __restrict__ score) {
  int wave = (blockIdx.x * blockDim.x + threadIdx.x) >> 5;
  int lane = threadIdx.x & 31;
  if (wave >= N_TRIP) return;
  int t0 = trip[3 * wave], t1 = trip[3 * wave + 1], t2 = trip[3 * wave + 2];
  const float* h = ent + (size_t)t0 * DIM;
  const float* t = ent + (size_t)t1 * DIM;
  const float* r = rel + (size_t)t2 * DIM;
  const float phase_k = PI_F / REL_RANGE_F;
  float acc = 0.0f;
  for (int k = lane; k < DIM / 2; k += 32) {
    float re_h = h[k], im_h = h[k + 64];
    float re_t = t[k], im_t = t[k + 64];
    float sr, cr;
    sincosf(r[k] * phase_k, &sr, &cr);
    float re_s = cr * re_t + sr * im_t - re_h;
    float im_s = cr * im_t - sr * re_t - im_h;
    acc += sqrtf(re_s * re_s + im_s * im_s);
  }
  for (int off = 16; off > 0; off >>= 1) acc += __shfl_xor(acc, off, 32);
  if (lane == 0) score[wave] = MARGIN_F - acc;
}

extern "C" void kernel_launch(void* const* d_in, const int* in_sizes, int n_in,
                              void* d_out, int out_size, void* d_ws, size_t ws_size,
                              hipStream_t stream) {
  const int*   trip   = (const int*)  d_in[0];
  const float* ent    = (const float*)d_in[1];
  const float* rel    = (const float*)d_in[2];
  const float* W_a    = (const float*)d_in[3];
  const float* b_a    = (const float*)d_in[4];
  const float* W_a2   = (const float*)d_in[5];
  const float* b_a2   = (const float*)d_in[6];
  const float* gamma0 = (const float*)d_in[7];
  const float* beta0  = (const float*)d_in[8];
  const float* gamma1 = (const float*)d_in[9];
  const float* beta1  = (const float*)d_in[10];

  float* out = (float*)d_out;
  float* ws  = (float*)d_ws;

  float* ent_scale = ws + WS_ENT_SCALE;
  float* rel_scale = ws + WS_REL_SCALE;
  float* Sa = ws + WS_SA; float* Qa = ws + WS_QA; float* Qr = ws + WS_QR;
  float* Wb = ws + WS_WB; float* bias2 = ws + WS_BIAS2;
  float* sc1 = ws + WS_SC1; float* sc2 = ws + WS_SC2;
  float* a1 = ws + WS_A1; float* b1n = ws + WS_B1N;
  float* ebs = ws + WS_EBS; float* cnt = ws + WS_CNT;

  float* h_ent = out + OUT_HENT;
  float* h_rel = out + OUT_HREL;
  float* score = out + OUT_SCORE;

  // zero all accumulators (ws) and the atomic-scatter output regions
  zero_kernel<<<512, 256, 0, stream>>>(ws, WS_TOTAL);
  zero_kernel<<<4096, 256, 0, stream>>>(out, N_ENT * DIM + N_REL * DIM);

  scale_kernel<<<(N_ENT + N_REL + 7) / 8, 256, 0, stream>>>(ent, rel, ent_scale, rel_scale);
  stats0_kernel<<<512, 256, 0, stream>>>(trip, ent, rel, ent_scale, rel_scale, Sa, Qa, Qr, cnt);
  prep_kernel<<<1, 256, 0, stream>>>(W_a, b_a, gamma0, beta0, Sa, Qa, Qr, Wb, bias2);

  gemm_kernel<false><<<2048, 256, 0, stream>>>(trip, ent, rel, ent_scale, rel_scale,
                                               Wb, bias2, nullptr, nullptr, nullptr, nullptr,
                                               sc1, sc2, nullptr, nullptr, nullptr);
  bn1fin_kernel<<<1, 128, 0, stream>>>(sc1, sc2, gamma1, beta1, a1, b1n);
  gemm_kernel<true><<<2048, 256, 0, stream>>>(trip, ent, rel, ent_scale, rel_scale,
                                              Wb, bias2, a1, b1n, W_a2, b_a2,
                                              nullptr, nullptr, h_ent, ebs, h_rel);

  finalize_kernel<<<4096, 256, 0, stream>>>(h_ent, ebs, h_rel, cnt);
  score_kernel<<<(N_TRIP + 7) / 8, 256, 0, stream>>>(trip, ent, rel, score);
}